// SpatialAttention_85478439125884
// MI455X (gfx1250) — compile-verified
//
#include <hip/hip_runtime.h>
#include <math.h>

typedef __attribute__((ext_vector_type(2))) float v2f;
typedef __attribute__((ext_vector_type(8))) float v8f;

#define HW 65536            // 256*256
#define IMG_FLOATS (3 * HW) // floats per image
#define FEAT 192
#define ENC 16

__global__ __launch_bounds__(256) void spatial_attn_wmma_kernel(
    const float* __restrict__ x,
    const float* __restrict__ Wenc,   // [16][192]
    const float* __restrict__ benc,   // [16]
    const float* __restrict__ Wdec,   // [192][16]
    const float* __restrict__ bdec,   // [192]
    float* __restrict__ y)
{
    // LDS staging (even row strides -> 8B-aligned b64 reads, spread banks)
    __shared__ float sWe[16][194];     // W_enc rows, stride 194
    __shared__ float sWd[192][18];     // W_dec rows, stride 18
    __shared__ float sBe[16];
    __shared__ float sBd[192];
    __shared__ float sH[8][16][18];    // per-wave h, [patch][enc], stride 18

    const int tid = threadIdx.x;
    for (int idx = tid; idx < 16 * FEAT; idx += 256) sWe[idx / FEAT][idx % FEAT] = Wenc[idx];
    for (int idx = tid; idx < FEAT * 16; idx += 256) sWd[idx / 16][idx % 16]   = Wdec[idx];
    if (tid < 16)  sBe[tid] = benc[tid];
    if (tid < 192) sBd[tid] = bdec[tid];
    __syncthreads();

    const int wave = tid >> 5;
    const int lane = tid & 31;
    const int half = lane >> 4;     // 0: lanes 0-15, 1: lanes 16-31
    const int l16  = lane & 15;
    const int koff = half << 1;     // K-pair offset per WMMA f32 operand layout

    // 16 consecutive patches per wave: same image, same patch-row, j = j0..j0+15
    const int gp0 = blockIdx.x * 128 + wave * 16;
    const int img = gp0 >> 10;          // / (32*32)
    const int pi  = (gp0 >> 5) & 31;    // patch row
    const int j   = (gp0 & 31) + l16;   // this lane's patch column

    const float* xb = x + (long)img * IMG_FLOATS + pi * (8 * 256) + j * 8;

    // ---------------- Encoder: h[16e x 16p] = W_enc * xp ----------------
    v8f acc = {0.f, 0.f, 0.f, 0.f, 0.f, 0.f, 0.f, 0.f};
    #pragma unroll 8
    for (int t = 0; t < 48; ++t) {
        const int k  = 4 * t + koff;             // even feature index (pair k, k+1)
        const int c  = k >> 6;
        const int rr = (k >> 3) & 7;
        const int w  = k & 7;
        v2f bv = *(const v2f*)(xb + c * HW + rr * 256 + w);  // x pair (8B aligned)
        v2f av = *(const v2f*)(&sWe[l16][k]);                // W_enc[e=l16][k..k+1]
        acc = __builtin_amdgcn_wmma_f32_16x16x4_f32(
            false, av, false, bv, (short)0, acc, false, false);
    }

    // bias + ReLU, stash h transposed [patch][enc] for decoder B-operand
    #pragma unroll
    for (int r = 0; r < 8; ++r) {
        const int e = r + (half << 3);           // C/D layout: VGPR r -> M = r + 8*half
        sH[wave][l16][e] = fmaxf(acc[r] + sBe[e], 0.f);
    }
    __syncthreads();

    // ---------------- Decoder: out[192 x 16p] = W_dec * h ----------------
    float ov[12][8];
    #pragma unroll
    for (int m = 0; m < 12; ++m) {
        v8f d = {0.f, 0.f, 0.f, 0.f, 0.f, 0.f, 0.f, 0.f};
        #pragma unroll
        for (int t = 0; t < 4; ++t) {
            const int kk = 4 * t + koff;                       // enc-dim pair
            v2f av = *(const v2f*)(&sWd[m * 16 + l16][kk]);    // W_dec[row][kk..kk+1]
            v2f bv = *(const v2f*)(&sH[wave][l16][kk]);        // h[kk..kk+1][p=l16]
            d = __builtin_amdgcn_wmma_f32_16x16x4_f32(
                false, av, false, bv, (short)0, d, false, false);
        }
        #pragma unroll
        for (int r = 0; r < 8; ++r) {
            const int f = m * 16 + (half << 3) + r;
            ov[m][r] = fmaxf(d[r] + sBd[f], 0.f);
        }
    }

    // ---------------- Softmax over 192 features of this lane's patch -----
    // lane holds 96 values; partner lane (lane ^ 16) holds the other 96.
    float mx = -3.402823466e38f;
    #pragma unroll
    for (int m = 0; m < 12; ++m)
        #pragma unroll
        for (int r = 0; r < 8; ++r) mx = fmaxf(mx, ov[m][r]);
    mx = fmaxf(mx, __shfl_xor(mx, 16, 32));

    float s = 0.f;
    #pragma unroll
    for (int m = 0; m < 12; ++m)
        #pragma unroll
        for (int r = 0; r < 8; ++r) s += __expf(ov[m][r] - mx);
    s += __shfl_xor(s, 16, 32);
    const float inv = 1.0f / s;

    // ---------------- y = softmax(out) * out, scatter as float4 ----------
    // For tile m / half: f = 16m + 8*half + r  ->  w = r (8 consecutive floats
    // of one image row) -> two aligned float4 stores per tile.
    float* yb = y + (long)img * IMG_FLOATS + pi * (8 * 256) + j * 8;
    #pragma unroll
    for (int m = 0; m < 12; ++m) {
        const int f0 = m * 16 + (half << 3);
        const int c  = f0 >> 6;
        const int rr = (f0 >> 3) & 7;
        float v[8];
        #pragma unroll
        for (int r = 0; r < 8; ++r)
            v[r] = __expf(ov[m][r] - mx) * inv * ov[m][r];
        float* dst = yb + c * HW + rr * 256;
        *(float4*)(dst)     = make_float4(v[0], v[1], v[2], v[3]);
        *(float4*)(dst + 4) = make_float4(v[4], v[5], v[6], v[7]);
    }
}

extern "C" void kernel_launch(void* const* d_in, const int* in_sizes, int n_in,
                              void* d_out, int out_size, void* d_ws, size_t ws_size,
                              hipStream_t stream) {
    const float* x    = (const float*)d_in[0];
    const float* Wenc = (const float*)d_in[1];
    const float* benc = (const float*)d_in[2];
    const float* Wdec = (const float*)d_in[3];
    const float* bdec = (const float*)d_in[4];
    float* out = (float*)d_out;

    const int n_img   = in_sizes[0] / IMG_FLOATS;   // 128
    const int patches = n_img * 32 * 32;            // 131072
    const int blocks  = patches / 128;              // 128 patches per 256-thread block

    spatial_attn_wmma_kernel<<<blocks, 256, 0, stream>>>(x, Wenc, benc, Wdec, bdec, out);
}